// EdgeGNN_13013750907308
// MI455X (gfx1250) — compile-verified
//
#include <hip/hip_runtime.h>

// EdgeGNN (NNConv x2 + BN + MLP head) fully fused for gfx1250.
// Heavy math runs as V_WMMA_F32_16X16X4_F32 on 16-edge tiles:
//   msg_tile[16x32] = (h (x) x_src)[16 x K] @ W2flat[K x 32]   (K = 32*IN)
// so the [E,IN,32] per-edge weight tensor is never materialized.

#define NN      8192
#define NE      131072
#define HID     32
#define BN_EPS  1e-5f

typedef float v2f __attribute__((ext_vector_type(2)));
typedef float v8f __attribute__((ext_vector_type(8)));

__device__ __forceinline__ v8f wmma4_f32(v2f a, v2f b, v8f c) {
  // 8 args: (neg_a, A, neg_b, B, c_mod, C, reuse_a, reuse_b)
  return __builtin_amdgcn_wmma_f32_16x16x4_f32(false, a, false, b, (short)0, c,
                                               false, false);
}

__device__ __forceinline__ void atomic_add_f32(float* p, float v) {
  __hip_atomic_fetch_add(p, v, __ATOMIC_RELAXED, __HIP_MEMORY_SCOPE_AGENT);
}

__global__ void zero_f32(float* __restrict__ p, int n) {
  int i = blockIdx.x * blockDim.x + threadIdx.x;
  if (i < n) p[i] = 0.0f;
}

// One wave (32 lanes) per 16-edge tile; 4 waves per block.
template <int IN>
__global__ __launch_bounds__(128) void edge_nnconv(
    const float* __restrict__ xin,        // [NN, IN] node features in
    const long long* __restrict__ eidx,   // [2, NE]
    const float* __restrict__ ea,         // [NE, 16]
    const float* __restrict__ w1,         // [16, 32]
    const float* __restrict__ b1,         // [32]
    const float* __restrict__ w2,         // [32, IN*32]  (cols = i*32 + o)
    const float* __restrict__ b2,         // [IN*32]
    float* __restrict__ accum,            // [NN, 32] atomic sums
    float* __restrict__ cnt)              // [NN] (or nullptr)
{
  __shared__ float hbuf[4][16][HID + 1];
  __shared__ float xbuf[4][16][IN + 2];
  __shared__ int   dstb[4][16];
  __shared__ int   srcb[4][16];

  const int lane = threadIdx.x & 31;
  const int w    = threadIdx.x >> 5;
  const int row  = lane & 15;      // A row / B,D column 0..15 (lanes 16-31 alias rows via hi)
  const int hi   = lane >> 4;      // half-wave select
  const int kOff = hi * 2;         // K sub-offset inside each 4-wide K step

  const long long e0 = (long long)(blockIdx.x * 4 + w) * 16;

  // ---- stage edge indices ----
  if (lane < 16) {
    long long e = e0 + lane;
    srcb[w][lane] = (int)eidx[e];
    dstb[w][lane] = (int)eidx[NE + e];
  }
  __syncthreads();

  // ---- gather x_src rows into LDS (each half-wave covers half the row) ----
  {
    const long long s = (long long)srcb[w][row];
    #pragma unroll
    for (int j = 0; j < IN / 2; j += 4) {
      int c = hi * (IN / 2) + j;
      float4 v = *(const float4*)&xin[s * IN + c];
      xbuf[w][row][c + 0] = v.x; xbuf[w][row][c + 1] = v.y;
      xbuf[w][row][c + 2] = v.z; xbuf[w][row][c + 3] = v.w;
    }
  }

  // ---- edge MLP hidden: h = relu(ea_tile[16x16] @ w1[16x32] + b1) ----
  v8f hc0 = {0,0,0,0,0,0,0,0};
  v8f hc1 = {0,0,0,0,0,0,0,0};
  #pragma unroll
  for (int k0 = 0; k0 < 16; k0 += 4) {
    v2f a;
    a.x = ea[(e0 + row) * 16 + k0 + kOff];
    a.y = ea[(e0 + row) * 16 + k0 + kOff + 1];
    v2f b;
    b.x = w1[(k0 + kOff) * 32 + row];           // N-tile 0: cols 0..15
    b.y = w1[(k0 + kOff + 1) * 32 + row];
    hc0 = wmma4_f32(a, b, hc0);
    b.x = w1[(k0 + kOff) * 32 + 16 + row];      // N-tile 1: cols 16..31
    b.y = w1[(k0 + kOff + 1) * 32 + 16 + row];
    hc1 = wmma4_f32(a, b, hc1);
  }
  #pragma unroll
  for (int t = 0; t < 2; ++t) {
    int col = t * 16 + row;
    float bv = b1[col];
    #pragma unroll
    for (int r = 0; r < 8; ++r) {
      float v = (t ? hc1[r] : hc0[r]) + bv;
      hbuf[w][r + hi * 8][col] = fmaxf(v, 0.0f);
    }
  }
  __syncthreads();

  // ---- main contraction: msg = P[16 x IN*32] @ W2flat + x_src @ B2r ----
  // P[e, k*IN+i] = h[e,k] * x_src[e,i]; within each 4-wide K step, k is constant.
  v8f mc0 = {0,0,0,0,0,0,0,0};
  v8f mc1 = {0,0,0,0,0,0,0,0};
  const int K4 = IN * 8;                        // (32*IN)/4 K-steps
  for (int kk = 0; kk < K4; ++kk) {
    const int ki0 = kk * 4;
    const int k   = ki0 / IN;
    const int i0  = ki0 % IN;
    const float hv = hbuf[w][row][k];
    v2f a;
    a.x = hv * xbuf[w][row][i0 + kOff];
    a.y = hv * xbuf[w][row][i0 + kOff + 1];
    const float* w2r = &w2[(long long)k * (IN * 32) + (i0 + kOff) * 32];
    v2f b;
    b.x = w2r[row];       b.y = w2r[32 + row];
    mc0 = wmma4_f32(a, b, mc0);
    b.x = w2r[16 + row];  b.y = w2r[48 + row];
    mc1 = wmma4_f32(a, b, mc1);
  }
  // bias term: x_src[16xIN] @ B2r[INx32], B2r[i][o] = b2[i*32+o]
  #pragma unroll
  for (int i0 = 0; i0 < IN; i0 += 4) {
    v2f a;
    a.x = xbuf[w][row][i0 + kOff];
    a.y = xbuf[w][row][i0 + kOff + 1];
    const float* b2r = &b2[(i0 + kOff) * 32];
    v2f b;
    b.x = b2r[row];       b.y = b2r[32 + row];
    mc0 = wmma4_f32(a, b, mc0);
    b.x = b2r[16 + row];  b.y = b2r[48 + row];
    mc1 = wmma4_f32(a, b, mc1);
  }

  // ---- scatter-add messages to destination nodes ----
  #pragma unroll
  for (int r = 0; r < 8; ++r) {
    const int m = r + hi * 8;
    const long long d = (long long)dstb[w][m];
    atomic_add_f32(&accum[d * 32 + row],      mc0[r]);
    atomic_add_f32(&accum[d * 32 + 16 + row], mc1[r]);
  }
  if (cnt != nullptr && lane < 16) {
    atomic_add_f32(&cnt[dstb[w][lane]], 1.0f);
  }
}

__global__ void node_update1(const float* __restrict__ x,     // [NN,16]
                             const float* __restrict__ sum1,  // [NN,32]
                             const float* __restrict__ cnt,   // [NN]
                             const float* __restrict__ root,  // [16,32]
                             const float* __restrict__ bias,
                             const float* __restrict__ g,  const float* __restrict__ be,
                             const float* __restrict__ mu, const float* __restrict__ var,
                             float* __restrict__ h1)          // [NN,32]
{
  int n = blockIdx.x * blockDim.x + threadIdx.x;
  if (n >= NN) return;
  float inv = 1.0f / fmaxf(cnt[n], 1.0f);
  float xr[16];
  #pragma unroll
  for (int i = 0; i < 16; ++i) xr[i] = x[n * 16 + i];
  #pragma unroll
  for (int o = 0; o < 32; ++o) {
    float acc = sum1[n * 32 + o] * inv + bias[o];
    #pragma unroll
    for (int i = 0; i < 16; ++i) acc += xr[i] * root[i * 32 + o];
    float bn = g[o] * (acc - mu[o]) * rsqrtf(var[o] + BN_EPS) + be[o];
    h1[n * 32 + o] = fmaxf(bn, 0.0f);
  }
}

__global__ void node_update2(const float* __restrict__ h1,    // [NN,32]
                             const float* __restrict__ sum2, const float* __restrict__ cnt,
                             const float* __restrict__ root,  // [32,32]
                             const float* __restrict__ bias,
                             const float* __restrict__ g,  const float* __restrict__ be,
                             const float* __restrict__ mu, const float* __restrict__ var,
                             const float* __restrict__ l1w,   // [32,16]
                             const float* __restrict__ l1b,
                             const float* __restrict__ l2w,   // [16,8]
                             const float* __restrict__ l2b,
                             float* __restrict__ y)           // [NN,8]
{
  int n = blockIdx.x * blockDim.x + threadIdx.x;
  if (n >= NN) return;
  float inv = 1.0f / fmaxf(cnt[n], 1.0f);
  float hr[32];
  #pragma unroll
  for (int i = 0; i < 32; ++i) hr[i] = h1[n * 32 + i];
  float h2[32];
  #pragma unroll
  for (int o = 0; o < 32; ++o) {
    float acc = sum2[n * 32 + o] * inv + bias[o];
    #pragma unroll
    for (int i = 0; i < 32; ++i) acc += hr[i] * root[i * 32 + o];
    float bn = g[o] * (acc - mu[o]) * rsqrtf(var[o] + BN_EPS) + be[o];
    h2[o] = fmaxf(bn, 0.0f);
  }
  float t16[16];
  #pragma unroll
  for (int o = 0; o < 16; ++o) {
    float acc = l1b[o];
    #pragma unroll
    for (int i = 0; i < 32; ++i) acc += h2[i] * l1w[i * 16 + o];
    t16[o] = fmaxf(acc, 0.0f);
  }
  #pragma unroll
  for (int o = 0; o < 8; ++o) {
    float acc = l2b[o];
    #pragma unroll
    for (int i = 0; i < 16; ++i) acc += t16[i] * l2w[i * 8 + o];
    y[n * 8 + o] = acc;
  }
}

__global__ void gather_out(const float* __restrict__ y,          // [NN,8]
                           const long long* __restrict__ src,    // eidx row 0
                           float* __restrict__ out)              // [NE,8]
{
  int e = blockIdx.x * blockDim.x + threadIdx.x;
  if (e >= NE) return;
  long long s = src[e];
  float4 a = *(const float4*)&y[s * 8];
  float4 b = *(const float4*)&y[s * 8 + 4];
  *(float4*)&out[(long long)e * 8]     = a;
  *(float4*)&out[(long long)e * 8 + 4] = b;
}

extern "C" void kernel_launch(void* const* d_in, const int* in_sizes, int n_in,
                              void* d_out, int out_size, void* d_ws, size_t ws_size,
                              hipStream_t stream) {
  const float*     x      = (const float*)d_in[0];
  const long long* eidx   = (const long long*)d_in[1];
  const float*     ea     = (const float*)d_in[2];
  const float*     nn1_w1 = (const float*)d_in[3];
  const float*     nn1_b1 = (const float*)d_in[4];
  const float*     nn1_w2 = (const float*)d_in[5];
  const float*     nn1_b2 = (const float*)d_in[6];
  const float*     root1  = (const float*)d_in[7];
  const float*     bias1  = (const float*)d_in[8];
  const float*     bn1_g  = (const float*)d_in[9];
  const float*     bn1_b  = (const float*)d_in[10];
  const float*     bn1_m  = (const float*)d_in[11];
  const float*     bn1_v  = (const float*)d_in[12];
  const float*     nn2_w1 = (const float*)d_in[13];
  const float*     nn2_b1 = (const float*)d_in[14];
  const float*     nn2_w2 = (const float*)d_in[15];
  const float*     nn2_b2 = (const float*)d_in[16];
  const float*     root2  = (const float*)d_in[17];
  const float*     bias2  = (const float*)d_in[18];
  const float*     bn2_g  = (const float*)d_in[19];
  const float*     bn2_b  = (const float*)d_in[20];
  const float*     bn2_m  = (const float*)d_in[21];
  const float*     bn2_v  = (const float*)d_in[22];
  const float*     lin1_w = (const float*)d_in[23];
  const float*     lin1_b = (const float*)d_in[24];
  const float*     lin2_w = (const float*)d_in[25];
  const float*     lin2_b = (const float*)d_in[26];

  // workspace layout (floats): sum1[N*32] sum2[N*32] cnt[N] h1[N*32] y[N*8]
  float* ws   = (float*)d_ws;
  float* sum1 = ws;
  float* sum2 = sum1 + (size_t)NN * 32;
  float* cnt  = sum2 + (size_t)NN * 32;
  float* h1   = cnt  + (size_t)NN;
  float* yv   = h1   + (size_t)NN * 32;

  const int nzero = NN * 32 * 2 + NN;   // sum1 | sum2 | cnt are contiguous
  zero_f32<<<(nzero + 255) / 256, 256, 0, stream>>>(ws, nzero);

  // layer 1: x[N,16] -> sum1, cnt
  edge_nnconv<16><<<NE / 64, 128, 0, stream>>>(x, eidx, ea, nn1_w1, nn1_b1,
                                               nn1_w2, nn1_b2, sum1, cnt);
  node_update1<<<NN / 256, 256, 0, stream>>>(x, sum1, cnt, root1, bias1,
                                             bn1_g, bn1_b, bn1_m, bn1_v, h1);
  // layer 2: h1[N,32] -> sum2
  edge_nnconv<32><<<NE / 64, 128, 0, stream>>>(h1, eidx, ea, nn2_w1, nn2_b1,
                                               nn2_w2, nn2_b2, sum2, nullptr);
  node_update2<<<NN / 256, 256, 0, stream>>>(h1, sum2, cnt, root2, bias2,
                                             bn2_g, bn2_b, bn2_m, bn2_v,
                                             lin1_w, lin1_b, lin2_w, lin2_b, yv);
  gather_out<<<NE / 256, 256, 0, stream>>>(yv, eidx, (float*)d_out);
}